// ddpOptimizer_14980845928901
// MI455X (gfx1250) — compile-verified
//
#include <hip/hip_runtime.h>

// ---------------------------------------------------------------------------
// IPDDP backward Riccati recursion on MI455X (gfx1250, wave32, WMMA f32).
// Single persistent workgroup (the scan is strictly sequential), 512 threads
// = 16 wave32s on one WGP. f32 WMMA 16x16x4 for all dense matmuls; streaming
// NT loads + cache prefetch for the big second-order tensors; TDM
// (tensor_load_to_lds) stages the next step's small matrices into LDS while
// the solve/gain phases run, synchronized with s_wait_tensorcnt.
// ---------------------------------------------------------------------------

typedef __attribute__((ext_vector_type(2))) float v2f;
typedef __attribute__((ext_vector_type(8))) float v8f;
typedef unsigned int uint32x4 __attribute__((ext_vector_type(4)));
typedef int int32x4 __attribute__((ext_vector_type(4)));
typedef int int32x8 __attribute__((ext_vector_type(8)));

#define N_STEPS 4096
#define NX 32
#define NU 16
#define NC 8
#define NT 512

#define MU_C        1.0f
#define REG_SCALE_C 0.0f   // REG_EXP_BASE**REG - 1 with REG=0

// staged small-input LDS segment offsets (floats)
#define S_FX   0        // 32*32
#define S_FU   1024     // 32*16
#define S_QXX  1536     // 32*32
#define S_QXU  2560     // 32*16
#define S_QUU  3072     // 16*16
#define S_QX   3328     // 32
#define S_QU   3360     // 16
#define S_CX   3376     // 8*32
#define S_CU   3632     // 8*16
#define S_C    3760     // 8
#define S_Y    3768     // 8
#define S_S    3776     // 8
#define S_TOT  3784

#if __has_builtin(__builtin_amdgcn_tensor_load_to_lds)
#define HAVE_TDM 1
#else
#define HAVE_TDM 0
#endif

__device__ __forceinline__ v8f vzero8() {
  v8f z = {0.f,0.f,0.f,0.f,0.f,0.f,0.f,0.f};
  return z;
}

#if HAVE_TDM
// 1-D contiguous TDM copy: n f32 elements global -> LDS.
// D# packing per CDNA5 ISA 8.3/8.4: group0 = {flags, lds_addr, global_addr,
// addr_hi|type=2}; group1 = {wg_mask/data_size, tensor dims, tile dims,
// strides}. 1-D tile: tensor_dim0 = tile_dim0 = n, tensor_dim1 = tile_dim1 = 1.
__device__ __forceinline__ void tdm_load_1d(const float* gptr,
                                            const float* ldsptr, int n) {
  const unsigned long long ga = reinterpret_cast<unsigned long long>(gptr);
  // flat shared-aperture pointer: low 32 bits == LDS byte offset (ISA 10.2)
  const unsigned int lo =
      (unsigned int)reinterpret_cast<unsigned long long>(ldsptr);
  const unsigned int un = (unsigned int)n;
  uint32x4 g0;
  g0[0] = 1u;                                            // count=1, user mode
  g0[1] = lo;                                            // lds_addr
  g0[2] = (unsigned int)(ga & 0xffffffffull);            // global_addr[31:0]
  g0[3] = (unsigned int)((ga >> 32) & 0x01ffffffull) | (2u << 30); // hi|type=2
  int32x8 g1;
  g1[0] = 0x20000;                   // workgroup_mask=0, data_size=2 (4 bytes)
  g1[1] = (int)(un << 16);           // tensor_dim0[15:0] (bits 63:48)
  g1[2] = (int)((un >> 16) | (1u << 16)); // tensor_dim0[31:16] | tensor_dim1=1
  g1[3] = (int)(un << 16);           // tensor_dim1 hi=0 | tile_dim0 = n
  g1[4] = 1;                         // tile_dim1=1, tile_dim2=0
  g1[5] = (int)un;                   // tensor_dim0_stride[31:0] = n
  g1[6] = 0;                         // stride0 hi, stride1 lo
  g1[7] = 0;                         // stride1 hi
  int32x4 g2 = {0, 0, 0, 0};
  int32x4 g3 = {0, 0, 0, 0};
#if defined(__clang_major__) && (__clang_major__ >= 23)
  int32x8 gz = {0, 0, 0, 0, 0, 0, 0, 0};
  __builtin_amdgcn_tensor_load_to_lds(g0, g1, g2, g3, gz, 0);
#else
  __builtin_amdgcn_tensor_load_to_lds(g0, g1, g2, g3, 0);
#endif
}
#endif  // HAVE_TDM

// acc += op(A) * B for one 16x16 tile, K in multiples of 4, f32 WMMA.
// A: if TRANSA, A-operand element (m,k) read as Astore[k*lda + m]. Otherwise
// Astore[m*lda+k]. B: K x 16 row-major, element (k,n) at B[k*ldb + n].
// VGPR layouts per CDNA5 ISA 7.12.2 (32-bit A 16x4, 32-bit B 4x16, C/D 16x16).
template <bool TRANSA>
__device__ __forceinline__ v8f wmma_mm(const float* __restrict__ A, int lda,
                                       const float* __restrict__ B, int ldb,
                                       int K, v8f acc) {
  const int lane = threadIdx.x & 31;
  const int mr = lane & 15;      // row of A / col of B handled by this lane
  const int h  = lane >> 4;      // half-wave selects K sub-pair
  for (int k0 = 0; k0 < K; k0 += 4) {
    const int ka = k0 + 2 * h;
    v2f a, b;
    if (TRANSA) {
      a.x = A[ka * lda + mr];
      a.y = A[(ka + 1) * lda + mr];
    } else {
      a.x = A[mr * lda + ka];
      a.y = A[mr * lda + ka + 1];
    }
    b.x = B[ka * ldb + mr];
    b.y = B[(ka + 1) * ldb + mr];
    acc = __builtin_amdgcn_wmma_f32_16x16x4_f32(false, a, false, b,
                                                (short)0, acc, false, false);
  }
  return acc;
}

__device__ __forceinline__ void wmma_store(float* __restrict__ C, int ldc, v8f acc) {
  const int lane = threadIdx.x & 31;
  const int col = lane & 15;
  const int h   = lane >> 4;
#pragma unroll
  for (int r = 0; r < 8; ++r) C[(r + 8 * h) * ldc + col] = acc[r];
}

__device__ __forceinline__ v8f wmma_loadC(const float* __restrict__ C, int ldc) {
  v8f acc;
  const int lane = threadIdx.x & 31;
  const int col = lane & 15;
  const int h   = lane >> 4;
#pragma unroll
  for (int r = 0; r < 8; ++r) acc[r] = C[(r + 8 * h) * ldc + col];
  return acc;
}

// flat staged-input index -> global address for a given step (fallback path)
__device__ __forceinline__ const float* seg_ptr(
    int f, int step,
    const float* fx, const float* fu, const float* qxx, const float* qxu,
    const float* quu, const float* qx, const float* qu, const float* cx,
    const float* cu, const float* c, const float* y, const float* s) {
  if (f < S_FU)  return fx  + (size_t)step * 1024 + f;
  if (f < S_QXX) return fu  + (size_t)step * 512  + (f - S_FU);
  if (f < S_QXU) return qxx + (size_t)step * 1024 + (f - S_QXX);
  if (f < S_QUU) return qxu + (size_t)step * 512  + (f - S_QXU);
  if (f < S_QX)  return quu + (size_t)step * 256  + (f - S_QUU);
  if (f < S_QU)  return qx  + (size_t)step * 32   + (f - S_QX);
  if (f < S_CX)  return qu  + (size_t)step * 16   + (f - S_QU);
  if (f < S_CU)  return cx  + (size_t)step * 256  + (f - S_CX);
  if (f < S_C)   return cu  + (size_t)step * 128  + (f - S_CU);
  if (f < S_Y)   return c   + (size_t)step * 8    + (f - S_C);
  if (f < S_S)   return y   + (size_t)step * 8    + (f - S_Y);
  return s + (size_t)step * 8 + (f - S_S);
}

__global__ __launch_bounds__(NT, 1)
void ipddp_backward_kernel(
    const float* __restrict__ g_fx,  const float* __restrict__ g_fu,
    const float* __restrict__ g_fxx, const float* __restrict__ g_fxu,
    const float* __restrict__ g_fuu, const float* __restrict__ g_qx,
    const float* __restrict__ g_qu,  const float* __restrict__ g_qxx,
    const float* __restrict__ g_qxu, const float* __restrict__ g_quu,
    const float* __restrict__ g_cx,  const float* __restrict__ g_cu,
    const float* __restrict__ g_c,   const float* __restrict__ g_y,
    const float* __restrict__ g_s,   const float* __restrict__ g_px,
    const float* __restrict__ g_pxx, float* __restrict__ g_out) {
  const int tid = threadIdx.x;
  const int wv  = tid >> 5;

  // carry
  __shared__ float ls_Vx[NX];
  __shared__ float ls_Vxx[NX * NX];
  // staged small inputs (one step; refilled by TDM after its last reader P4)
  __shared__ float ls_in[S_TOT];
  // intermediates
  __shared__ float ls_fxTV[NX * NX];     // fx^T Vxx
  __shared__ float ls_fuTV[NU * NX];     // fu^T Vxx
  __shared__ float ls_Qxx[NX * NX];      // becomes Qxx_h
  __shared__ float ls_Qxu[NX * NU];
  __shared__ float ls_Quu[NU * NU];
  __shared__ float ls_cxSYcx[NX * NX];
  __shared__ float ls_cuSYcx[NU * NX];
  __shared__ float ls_cuSYcu[NU * NU];
  __shared__ float ls_SYcx[NC * NX];
  __shared__ float ls_SYcu[NC * NU];
  __shared__ float ls_Qx[NX], ls_Qu[NU], ls_Qxh[NX], ls_Quh[NU];
  __shared__ float ls_yr[NC];            // yinv * rhat
  __shared__ float ls_M[NU * 49];        // [A | Qu_h | tempQux] augmented
  __shared__ float ls_colk[NU];
  __shared__ float ls_Quuh[NU * NU];
  __shared__ float ls_Qxuh[NX * NU];     // tempQux^T  (32x16)
  __shared__ float ls_ku[NU];
  __shared__ float ls_Ku[NU * NX];
  __shared__ float ls_w16[NU];
  __shared__ float ls_T1[NU * NX];       // Quu_h @ Ku
  __shared__ float ls_G[NX * NX];        // Qxu_h @ Ku
  __shared__ float ls_Vxxn[NX * NX];

  // stage one step's small inputs into ls_in (TDM if available)
  auto stage_small = [&](int s) {
#if HAVE_TDM
    if (wv == 15) {  // one wave issues the DMA; TENSORcnt is per-wave
      tdm_load_1d(g_fx  + (size_t)s * 1024, ls_in + S_FX,  1024);
      tdm_load_1d(g_fu  + (size_t)s * 512,  ls_in + S_FU,  512);
      tdm_load_1d(g_qxx + (size_t)s * 1024, ls_in + S_QXX, 1024);
      tdm_load_1d(g_qxu + (size_t)s * 512,  ls_in + S_QXU, 512);
      tdm_load_1d(g_quu + (size_t)s * 256,  ls_in + S_QUU, 256);
      tdm_load_1d(g_qx  + (size_t)s * 32,   ls_in + S_QX,  32);
      tdm_load_1d(g_qu  + (size_t)s * 16,   ls_in + S_QU,  16);
      tdm_load_1d(g_cx  + (size_t)s * 256,  ls_in + S_CX,  256);
      tdm_load_1d(g_cu  + (size_t)s * 128,  ls_in + S_CU,  128);
      tdm_load_1d(g_c   + (size_t)s * 8,    ls_in + S_C,   8);
      tdm_load_1d(g_y   + (size_t)s * 8,    ls_in + S_Y,   8);
      tdm_load_1d(g_s   + (size_t)s * 8,    ls_in + S_S,   8);
    }
#else
    for (int f = tid; f < S_TOT; f += NT)
      ls_in[f] = *seg_ptr(f, s, g_fx, g_fu, g_qxx, g_qxu, g_quu,
                          g_qx, g_qu, g_cx, g_cu, g_c, g_y, g_s);
#endif
  };
  auto stage_wait = [&]() {
#if HAVE_TDM
    if (wv == 15) __builtin_amdgcn_s_wait_tensorcnt(0);
#endif
  };

  // ---- init carry: Vx = px, Vxx = pxx; stage first step's smalls ----
  for (int i = tid; i < NX; i += NT) ls_Vx[i] = g_px[i];
  for (int i = tid; i < NX * NX; i += NT) ls_Vxx[i] = g_pxx[i];
  stage_small(N_STEPS - 1);
  stage_wait();
  __syncthreads();

  for (int step = N_STEPS - 1; step >= 0; --step) {
    // ---- P1: slack/scalar terms, Qx/Qu, and streamed vx-contractions ----
    if (tid < NC) {
      const float yi = ls_in[S_Y + tid], si = ls_in[S_S + tid], ci = ls_in[S_C + tid];
      const float r    = si * yi - MU_C;
      const float rhat = si * (ci + yi) - r;
      const float yinv = 1.0f / yi;
      ls_yr[tid] = yinv * rhat;
      const float sy = si * yinv;
      for (int j = 0; j < NX; ++j) ls_SYcx[tid * NX + j] = ls_in[S_CX + tid * NX + j] * sy;
      for (int j = 0; j < NU; ++j) ls_SYcu[tid * NU + j] = ls_in[S_CU + tid * NU + j] * sy;
    } else if (tid < NC + NX) {
      const int j = tid - NC;                 // Qx = qx + cx^T s + fx^T Vx
      float a = ls_in[S_QX + j];
      for (int k = 0; k < NC; ++k) a = fmaf(ls_in[S_CX + k * NX + j], ls_in[S_S + k], a);
      for (int k = 0; k < NX; ++k) a = fmaf(ls_in[S_FX + k * NX + j], ls_Vx[k], a);
      ls_Qx[j] = a;
    } else if (tid < NC + NX + NU) {
      const int i = tid - NC - NX;            // Qu = qu + cu^T s + fu^T Vx
      float a = ls_in[S_QU + i];
      for (int k = 0; k < NC; ++k) a = fmaf(ls_in[S_CU + k * NU + i], ls_in[S_S + k], a);
      for (int k = 0; k < NX; ++k) a = fmaf(ls_in[S_FU + k * NU + i], ls_Vx[k], a);
      ls_Qu[i] = a;
    }
    {
      // sum_k Vx[k] * {fxx,fxu,fuu}[k] streamed from HBM (non-temporal)
      const float* pfxx = g_fxx + (size_t)step * 32768;
      const float* pfxu = g_fxu + (size_t)step * 16384;
      const float* pfuu = g_fuu + (size_t)step * 8192;
      float a0 = 0.f, a1 = 0.f, a2 = 0.f;
#pragma unroll 8
      for (int k = 0; k < NX; ++k) {
        const float v = ls_Vx[k];
        a0 = fmaf(v, __builtin_nontemporal_load(pfxx + k * 1024 + tid), a0);
        a1 = fmaf(v, __builtin_nontemporal_load(pfxx + k * 1024 + 512 + tid), a1);
        a2 = fmaf(v, __builtin_nontemporal_load(pfxu + k * 512 + tid), a2);
      }
      ls_Qxx[tid]       = ls_in[S_QXX + tid] + a0;
      ls_Qxx[tid + 512] = ls_in[S_QXX + tid + 512] + a1;
      ls_Qxu[tid]       = ls_in[S_QXU + tid] + a2;
      if (tid < 256) {
        float a3 = 0.f;
#pragma unroll 8
        for (int k = 0; k < NX; ++k)
          a3 = fmaf(ls_Vx[k], __builtin_nontemporal_load(pfuu + k * 256 + tid), a3);
        ls_Quu[tid] = ls_in[S_QUU + tid] + a3;
      }
    }
    __syncthreads();

    // ---- P2 (WMMA): fx^T Vxx, fu^T Vxx, cu^T SYcu, cx^T SYcx, cu^T SYcx ----
    if (wv < 4) {                                   // fxTV: 32x32, K=32
      const int bi = wv >> 1, bj = wv & 1;
      v8f acc = vzero8();
      acc = wmma_mm<true>(ls_in + S_FX + 16 * bi, NX, ls_Vxx + 16 * bj, NX, NX, acc);
      wmma_store(ls_fxTV + 16 * bi * NX + 16 * bj, NX, acc);
    } else if (wv < 6) {                            // fuTV: 16x32, K=32
      const int bj = wv - 4;
      v8f acc = vzero8();
      acc = wmma_mm<true>(ls_in + S_FU, NU, ls_Vxx + 16 * bj, NX, NX, acc);
      wmma_store(ls_fuTV + 16 * bj, NX, acc);
    } else if (wv == 6) {                           // cu^T SYcu: 16x16, K=8
      v8f acc = vzero8();
      acc = wmma_mm<true>(ls_in + S_CU, NU, ls_SYcu, NU, NC, acc);
      wmma_store(ls_cuSYcu, NU, acc);
    } else if (wv < 11) {                           // cx^T SYcx: 32x32, K=8
      const int t = wv - 7, bi = t >> 1, bj = t & 1;
      v8f acc = vzero8();
      acc = wmma_mm<true>(ls_in + S_CX + 16 * bi, NX, ls_SYcx + 16 * bj, NX, NC, acc);
      wmma_store(ls_cxSYcx + 16 * bi * NX + 16 * bj, NX, acc);
    } else if (wv < 13) {                           // cu^T SYcx: 16x32, K=8
      const int bj = wv - 11;
      v8f acc = vzero8();
      acc = wmma_mm<true>(ls_in + S_CU, NU, ls_SYcx + 16 * bj, NX, NC, acc);
      wmma_store(ls_cuSYcx + 16 * bj, NX, acc);
    } else {                                        // Qx_h, Qu_h (VALU)
      const int t = tid - 13 * 32;
      if (t < NX) {
        float a = ls_Qx[t];
        for (int k = 0; k < NC; ++k) a = fmaf(ls_in[S_CX + k * NX + t], ls_yr[k], a);
        ls_Qxh[t] = a;
      } else if (t < NX + NU) {
        const int i = t - NX;
        float a = ls_Qu[i];
        for (int k = 0; k < NC; ++k) a = fmaf(ls_in[S_CU + k * NU + i], ls_yr[k], a);
        ls_Quh[i] = a;
      }
    }
    __syncthreads();

    // ---- P3 (WMMA): Qxx_h, Qxu, Quu ----
    if (wv < 4) {                                   // Qxx_h = Qxx + fxTV@fx + cxSYcx
      const int bi = wv >> 1, bj = wv & 1;
      float* base = ls_Qxx + 16 * bi * NX + 16 * bj;
      const float* add = ls_cxSYcx + 16 * bi * NX + 16 * bj;
      v8f acc;
      {
        const int lane = tid & 31, col = lane & 15, h = lane >> 4;
#pragma unroll
        for (int r = 0; r < 8; ++r)
          acc[r] = base[(r + 8 * h) * NX + col] + add[(r + 8 * h) * NX + col];
      }
      acc = wmma_mm<false>(ls_fxTV + 16 * bi * NX, NX, ls_in + S_FX + 16 * bj, NX, NX, acc);
      wmma_store(base, NX, acc);
    } else if (wv < 6) {                            // Qxu += fxTV @ fu
      const int bi = wv - 4;
      float* base = ls_Qxu + 16 * bi * NU;
      v8f acc = wmma_loadC(base, NU);
      acc = wmma_mm<false>(ls_fxTV + 16 * bi * NX, NX, ls_in + S_FU, NU, NX, acc);
      wmma_store(base, NU, acc);
    } else if (wv == 6) {                           // Quu += fuTV @ fu
      v8f acc = wmma_loadC(ls_Quu, NU);
      acc = wmma_mm<false>(ls_fuTV, NX, ls_in + S_FU, NU, NX, acc);
      wmma_store(ls_Quu, NU, acc);
    }
    __syncthreads();

    // ---- P4: build augmented system M = [A | Qu_h | tempQux], save Quu_h/Qxu_h
    for (int idx = tid; idx < NU * 49; idx += NT) {
      const int i = idx / 49, j = idx - 49 * i;
      float v;
      if (j < 16) {
        const float quuh = 0.5f * (ls_Quu[i * NU + j] + ls_Quu[j * NU + i]) +
                           ls_cuSYcu[i * NU + j];
        ls_Quuh[i * NU + j] = quuh;
        v = quuh + REG_SCALE_C * ls_in[S_QUU + i * NU + j];
      } else if (j == 16) {
        v = ls_Quh[i];
      } else {
        const int jc = j - 17;
        v = ls_Qxu[jc * NU + i] + ls_cuSYcx[i * NX + jc];
        ls_Qxuh[jc * NU + i] = v;
      }
      ls_M[idx] = v;
    }
    __syncthreads();

    // ls_in's last reader was P4: start TDM refill for the next step now; it
    // runs under P5..P8 and is waited on before the next iteration.
    if (step > 0) stage_small(step - 1);

    // ---- P5: Gauss-Jordan solve (A well conditioned: 2I + O(0.1)) ----
    for (int k = 0; k < NU; ++k) {
      if (tid < NU) ls_colk[tid] = ls_M[tid * 49 + k];
      __syncthreads();
      const float pinv = 1.0f / ls_colk[k];
      if (tid < 49) ls_M[k * 49 + tid] *= pinv;
      __syncthreads();
      for (int idx = tid; idx < NU * 49; idx += NT) {
        const int i = idx / 49, j = idx - 49 * i;
        if (i != k) ls_M[idx] = fmaf(-ls_colk[i], ls_M[k * 49 + j], ls_M[idx]);
      }
      __syncthreads();
    }
    // kK = -solution; emit output and keep ku/Ku in LDS
    for (int idx = tid; idx < NU * 33; idx += NT) {
      const int i = idx / 33, col = idx - 33 * i;
      const float v = -ls_M[i * 49 + 16 + col];
      g_out[(size_t)step * (NU * 33) + idx] = v;
      if (col == 0) ls_ku[i] = v;
      else ls_Ku[i * NX + (col - 1)] = v;
    }
    // prefetch next step's big tensors into cache while pipeline drains
    if (step > 0) {
      const char* t0 = (const char*)(g_fxx + (size_t)(step - 1) * 32768);
      const char* t1 = (const char*)(g_fxu + (size_t)(step - 1) * 16384);
      const char* t2 = (const char*)(g_fuu + (size_t)(step - 1) * 8192);
      for (int l = tid; l < 1024; l += NT) __builtin_prefetch(t0 + l * 128, 0, 1);
      for (int l = tid; l < 512;  l += NT) __builtin_prefetch(t1 + l * 128, 0, 1);
      if (tid < 256) __builtin_prefetch(t2 + tid * 128, 0, 1);
    }
    __syncthreads();

    // ---- P6 (WMMA): T1 = Quu_h@Ku, G = Qxu_h@Ku; w16 = Qu_h + Quu_h@ku ----
    if (wv < 2) {
      const int bj = wv;
      v8f acc = vzero8();
      acc = wmma_mm<false>(ls_Quuh, NU, ls_Ku + 16 * bj, NX, NU, acc);
      wmma_store(ls_T1 + 16 * bj, NX, acc);
    } else if (wv < 6) {
      const int t = wv - 2, bi = t >> 1, bj = t & 1;
      v8f acc = vzero8();
      acc = wmma_mm<false>(ls_Qxuh + 16 * bi * NU, NU, ls_Ku + 16 * bj, NX, NU, acc);
      wmma_store(ls_G + 16 * bi * NX + 16 * bj, NX, acc);
    } else if (wv == 6) {
      const int i = tid & 31;
      if (i < NU) {
        float a = ls_Quh[i];
        for (int k = 0; k < NU; ++k) a = fmaf(ls_Quuh[i * NU + k], ls_ku[k], a);
        ls_w16[i] = a;
      }
    }
    __syncthreads();

    // ---- P7 (WMMA): Vxx_n = Qxx_h + G + G^T + Ku^T @ T1;  Vx_n (VALU) ----
    if (wv < 4) {
      const int bi = wv >> 1, bj = wv & 1;
      v8f acc;
      {
        const int lane = tid & 31, col = lane & 15, h = lane >> 4;
#pragma unroll
        for (int r = 0; r < 8; ++r) {
          const int ri = 16 * bi + r + 8 * h, ci = 16 * bj + col;
          acc[r] = ls_Qxx[ri * NX + ci] + ls_G[ri * NX + ci] + ls_G[ci * NX + ri];
        }
      }
      acc = wmma_mm<true>(ls_Ku + 16 * bi, NX, ls_T1 + 16 * bj, NX, NU, acc);
      wmma_store(ls_Vxxn + 16 * bi * NX + 16 * bj, NX, acc);
    } else if (wv == 4) {
      const int j = tid & 31;  // Vx_n = Qx_h + Ku^T(Qu_h + Quu_h ku) + Qxu_h ku
      float a = ls_Qxh[j];
      for (int i = 0; i < NU; ++i) a = fmaf(ls_Ku[i * NX + j], ls_w16[i], a);
      for (int i = 0; i < NU; ++i) a = fmaf(ls_Qxuh[j * NU + i], ls_ku[i], a);
      ls_Vx[j] = a;
    }
    __syncthreads();

    // ---- P8: symmetrize Vxx into the carry; wait for TDM refill ----
    for (int idx = tid; idx < NX * NX; idx += NT) {
      const int i = idx >> 5, j = idx & 31;
      ls_Vxx[idx] = 0.5f * (ls_Vxxn[i * NX + j] + ls_Vxxn[j * NX + i]);
    }
    if (step > 0) stage_wait();
    __syncthreads();
  }
}

extern "C" void kernel_launch(void* const* d_in, const int* in_sizes, int n_in,
                              void* d_out, int out_size, void* d_ws, size_t ws_size,
                              hipStream_t stream) {
  (void)in_sizes; (void)n_in; (void)out_size; (void)d_ws; (void)ws_size;
  const float* fx  = (const float*)d_in[0];
  const float* fu  = (const float*)d_in[1];
  const float* fxx = (const float*)d_in[2];
  const float* fxu = (const float*)d_in[3];
  const float* fuu = (const float*)d_in[4];
  const float* qx  = (const float*)d_in[5];
  const float* qu  = (const float*)d_in[6];
  const float* qxx = (const float*)d_in[7];
  const float* qxu = (const float*)d_in[8];
  const float* quu = (const float*)d_in[9];
  const float* cx  = (const float*)d_in[10];
  const float* cu  = (const float*)d_in[11];
  const float* c   = (const float*)d_in[12];
  const float* y   = (const float*)d_in[13];
  const float* s   = (const float*)d_in[14];
  const float* px  = (const float*)d_in[15];
  const float* pxx = (const float*)d_in[16];
  float* out = (float*)d_out;

  ipddp_backward_kernel<<<dim3(1), dim3(NT), 0, stream>>>(
      fx, fu, fxx, fxu, fuu, qx, qu, qxx, qxu, quu, cx, cu, c, y, s, px, pxx, out);
}